// YoloV3Loss_80281528697676
// MI455X (gfx1250) — compile-verified
//
#include <hip/hip_runtime.h>
#include <hip/hip_bf16.h>

// ---------------- constants ----------------
#define NUM_CLASSES 80
#define IGNORE_THR  0.3f
#define L_XY    0.05f
#define L_WH    0.05f
#define L_CLASS 0.5f
#define L_CONF  1.0f
#define L_OBJ   1.0f
#define L_NOOBJ 1.0f
#define BCE_EPS 1e-7f

__device__ const float ANCH[9][2] = {
    {116.0f/416.0f,  90.0f/416.0f},
    {156.0f/416.0f, 198.0f/416.0f},
    {373.0f/416.0f, 326.0f/416.0f},
    { 30.0f/416.0f,  61.0f/416.0f},
    { 62.0f/416.0f,  45.0f/416.0f},
    { 59.0f/416.0f, 119.0f/416.0f},
    { 10.0f/416.0f,  13.0f/416.0f},
    { 16.0f/416.0f,  30.0f/416.0f},
    { 33.0f/416.0f,  23.0f/416.0f}};

typedef __attribute__((ext_vector_type(2))) float v2f;
typedef __attribute__((ext_vector_type(8))) float v8f;

__device__ __forceinline__ float sigm(float x) {
    return 1.0f / (1.0f + __expf(-x));
}
__device__ __forceinline__ float bce(float p, float t) {
    p = fminf(fmaxf(p, BCE_EPS), 1.0f - BCE_EPS);
    return -(t * __logf(p) + (1.0f - t) * __logf(1.0f - p));
}

// ---------------- kernel 1: zero workspace + output ----------------
__global__ void init_ws_kernel(float* __restrict__ W, int nfloats,
                               float* __restrict__ out) {
    int i = blockIdx.x * blockDim.x + threadIdx.x;
    if (i < nfloats) W[i] = 0.0f;
    if (i == 0) out[0] = 0.0f;
}

// ---------------- kernel 2: scatter targets ----------------
// Target layout per pyramid (cells = B*3*S*S):
//   field 0..5 : tx, ty, tw, th, w, h     (float)
//   field 6    : obj                       (float)
//   field 7    : class label               (int, aliased)
__global__ void build_targets_kernel(const float* __restrict__ boxes,
                                     const int*   __restrict__ labels,
                                     float* __restrict__ T13,
                                     float* __restrict__ T26,
                                     float* __restrict__ T52,
                                     int B, int N) {
    int i = blockIdx.x * blockDim.x + threadIdx.x;
    if (i >= B * N) return;
    int b = i / N;
    const float* bx = boxes + (size_t)i * 4;
    float cx = bx[0], cy = bx[1], w = bx[2], h = bx[3];

    int best = 0; float best_r = -1.0f;
#pragma unroll
    for (int k = 0; k < 9; ++k) {
        float aw = ANCH[k][0], ah = ANCH[k][1];
        float inter = fminf(w, aw) * fminf(h, ah);
        float uni   = w * h + aw * ah - inter;
        float r = inter / uni;
        if (r > best_r) { best_r = r; best = k; }
    }
    int pyr = best / 3, a = best % 3;
    int S = (pyr == 0) ? 13 : ((pyr == 1) ? 26 : 52);
    float* T = (pyr == 0) ? T13 : ((pyr == 1) ? T26 : T52);
    int cells = B * 3 * S * S;

    int gx = (int)floorf(cx * (float)S);
    int gy = (int)floorf(cy * (float)S);
    if (gx < 0 || gx >= S || gy < 0 || gy >= S) return;   // drop (OOB scatter)
    int idx = ((b * 3 + a) * S + gy) * S + gx;

    T[0 * cells + idx] = cx * (float)S - (float)gx;
    T[1 * cells + idx] = cy * (float)S - (float)gy;
    T[2 * cells + idx] = __logf(w / ANCH[best][0]);
    T[3 * cells + idx] = __logf(h / ANCH[best][1]);
    T[4 * cells + idx] = w;
    T[5 * cells + idx] = h;
    T[6 * cells + idx] = 1.0f;
    ((int*)T)[7 * cells + idx] = labels[i];
}

// ---------------- kernel 3: per-pyramid loss ----------------
__global__ void pyramid_loss_kernel(const float* __restrict__ pred,
                                    const float* __restrict__ boxes,
                                    const float* __restrict__ T,
                                    float* __restrict__ out,
                                    int S, int pyr, int B, int N,
                                    float invB) {
    const int SS    = S * S;
    const int CPB   = 3 * SS;          // cells per batch image
    const int cells = B * CPB;         // field stride in T
    const int b = blockIdx.x;
    const int c = blockIdx.y * blockDim.x + threadIdx.x;

    __shared__ float shb[256];         // up to 64 boxes * 4
    for (int i = threadIdx.x; i < N * 4; i += blockDim.x)
        shb[i] = boxes[(size_t)b * N * 4 + i];
    __syncthreads();

    float partial = 0.0f;
    if (c < CPB) {
        int a   = c / SS;
        int rem = c - a * SS;
        int gy  = rem / S;
        int gx  = rem - gy * S;

        const float* pc = pred + ((size_t)b * 255 + (size_t)a * 85) * SS
                               + (size_t)gy * S + gx;
        float p0 = pc[0];
        float p1 = pc[(size_t)SS];
        float p2 = pc[2 * (size_t)SS];
        float p3 = pc[3 * (size_t)SS];
        float p4 = pc[4 * (size_t)SS];

        float sx = sigm(p0), sy = sigm(p1), pconf = sigm(p4);
        float aW = ANCH[pyr * 3 + a][0], aH = ANCH[pyr * 3 + a][1];

        // decode to normalized box, max IoU vs all GT boxes (ignore mask)
        float invS = 1.0f / (float)S;
        float px = (sx + (float)gx) * invS;
        float py = (sy + (float)gy) * invS;
        float pw = __expf(p2) * aW;
        float ph = __expf(p3) * aH;
        float ax1 = px - 0.5f * pw, ax2 = px + 0.5f * pw;
        float ay1 = py - 0.5f * ph, ay2 = py + 0.5f * ph;
        float parea = (ax2 - ax1) * (ay2 - ay1);
        float miou = 0.0f;
        for (int n = 0; n < N; ++n) {
            float bcx = shb[4 * n + 0], bcy = shb[4 * n + 1];
            float bw  = shb[4 * n + 2], bh  = shb[4 * n + 3];
            float bx1 = bcx - 0.5f * bw, bx2 = bcx + 0.5f * bw;
            float by1 = bcy - 0.5f * bh, by2 = bcy + 0.5f * bh;
            float iw = fmaxf(fminf(ax2, bx2) - fmaxf(ax1, bx1), 0.0f);
            float ih = fmaxf(fminf(ay2, by2) - fmaxf(ay1, by1), 0.0f);
            float inter = iw * ih;
            float uni = parea + bw * bh - inter;
            miou = fmaxf(miou, inter / uni);
        }

        int idx = b * CPB + c;
        float obj   = T[6 * cells + idx];
        float noobj = (1.0f - obj) * ((miou <= IGNORE_THR) ? 1.0f : 0.0f);
        float cell  = L_CONF * bce(pconf, obj) * (L_OBJ * obj + L_NOOBJ * noobj);

        if (obj > 0.0f) {   // sparse path: class channels only at obj cells
            float tx = T[0 * cells + idx], ty = T[1 * cells + idx];
            float tw = T[2 * cells + idx], th = T[3 * cells + idx];
            float wb = T[4 * cells + idx], hb = T[5 * cells + idx];
            int  lab = ((const int*)T)[7 * cells + idx];
            float scale = 2.0f - wb * hb;

            float lxy = (bce(sx, tx) + bce(sy, ty)) * scale;
            float lwh = 0.5f * ((p2 - tw) * (p2 - tw) + (p3 - th) * (p3 - th)) * scale;

            __builtin_prefetch(pc + 5 * (size_t)SS, 0, 1);   // global_prefetch_b8
            float lcls = 0.0f;
            for (int k = 0; k < NUM_CLASSES; ++k) {
                float s = sigm(pc[(size_t)(5 + k) * SS]);
                lcls += bce(s, (k == lab) ? 1.0f : 0.0f);
            }
            cell += L_XY * lxy + L_WH * lwh + L_CLASS * lcls;
        }
        partial = cell;
    }

    // ---- wave reduction via V_WMMA_F32_16X16X4_F32 (B = ones) ----
    // A[16x4]: lane m (0..15) K=0 holds s_m, lanes 16..31 K=2 hold s_{m+16};
    // D[m][n] = s_m + s_{m+16}. Each lane sums its 8 D rows, then xor-16
    // pairs halves -> full 32-lane sum in every lane. Exact f32 adds.
    float hsum;
#if __has_builtin(__builtin_amdgcn_wmma_f32_16x16x4_f32)
    {
        v2f av; av[0] = partial; av[1] = 0.0f;
        v2f bv; bv[0] = 1.0f;    bv[1] = 1.0f;
        v8f cv = {};
        cv = __builtin_amdgcn_wmma_f32_16x16x4_f32(
                 false, av, false, bv, (short)0, cv, false, false);
        hsum = cv[0] + cv[1] + cv[2] + cv[3] + cv[4] + cv[5] + cv[6] + cv[7];
        hsum += __shfl_xor(hsum, 16, 32);
    }
#else
    {
        hsum = partial;
        for (int off = 16; off > 0; off >>= 1)
            hsum += __shfl_xor(hsum, off, 32);
    }
#endif
    if ((threadIdx.x & 31) == 0)
        atomicAdd(out, hsum * invB);
}

// ---------------- host launcher ----------------
extern "C" void kernel_launch(void* const* d_in, const int* in_sizes, int n_in,
                              void* d_out, int out_size, void* d_ws, size_t ws_size,
                              hipStream_t stream) {
    const float* p13    = (const float*)d_in[0];
    const float* p26    = (const float*)d_in[1];
    const float* p52    = (const float*)d_in[2];
    const float* boxes  = (const float*)d_in[3];
    const int*   labels = (const int*)d_in[4];
    float*       out    = (float*)d_out;

    const int B = in_sizes[0] / (255 * 13 * 13);
    const int N = in_sizes[4] / B;

    const int c13 = B * 3 * 13 * 13;
    const int c26 = B * 3 * 26 * 26;
    const int c52 = B * 3 * 52 * 52;

    float* W   = (float*)d_ws;
    float* T13 = W;
    float* T26 = T13 + (size_t)8 * c13;
    float* T52 = T26 + (size_t)8 * c26;
    const int totFloats = 8 * (c13 + c26 + c52);   // ~5.5 MB for B=16

    init_ws_kernel<<<(totFloats + 255) / 256, 256, 0, stream>>>(W, totFloats, out);

    const int nb = B * N;
    build_targets_kernel<<<(nb + 255) / 256, 256, 0, stream>>>(
        boxes, labels, T13, T26, T52, B, N);

    const float invB = 1.0f / (float)B;
    pyramid_loss_kernel<<<dim3(B, (3 * 13 * 13 + 255) / 256), 256, 0, stream>>>(
        p13, boxes, T13, out, 13, 0, B, N, invB);
    pyramid_loss_kernel<<<dim3(B, (3 * 26 * 26 + 255) / 256), 256, 0, stream>>>(
        p26, boxes, T26, out, 26, 1, B, N, invB);
    pyramid_loss_kernel<<<dim3(B, (3 * 52 * 52 + 255) / 256), 256, 0, stream>>>(
        p52, boxes, T52, out, 52, 2, B, N, invB);
}